// MultipleNegativesRankingLoss_43782896615450
// MI455X (gfx1250) — compile-verified
//
#include <hip/hip_runtime.h>
#include <math.h>

// Problem sizes (fixed by the reference)
constexpr int Qn = 32;    // queries
constexpr int Sn = 128;   // tokens per query
constexpr int En = 256;   // embedding dim
constexpr int Bn = 64;    // candidates
#define SCALE_F 20.0f

typedef __attribute__((ext_vector_type(2))) float v2f;
typedef __attribute__((ext_vector_type(8))) float v8f;

// ---------------- LDS layout (floats), padded strides to avoid bank conflicts
// A rows stride 260 (=256+4): 16 consecutive rows hit distinct banks.
constexpr int SA = 260;                 // anchors slice  128 x 256
constexpr int SC = 260;                 // candidates     64  x 256
constexpr int SD = 68;                  // dist matrix    128 x 64
constexpr int OFF_A = 0;
constexpr int OFF_C = OFF_A + Sn * SA;  // 33280
constexpr int OFF_D = OFF_C + Bn * SC;  // 49920
constexpr int OFF_M = OFF_D + Sn * SD;  // 58624  (mask as float, 128)
constexpr int OFF_R = OFF_M + Sn;       // 58752  (reduction scratch, 32)
constexpr int SMEM_FLOATS = OFF_R + 32; // 58784
constexpr int SMEM_BYTES  = SMEM_FLOATS * 4;  // 235136 B  (< 320 KB WGP LDS)

// ---------------- CDNA5 async global->LDS copy (ASYNCcnt-tracked DMA path) --
// Generic pointers to LDS carry the 32-bit LDS address in their low DWORD.
__device__ __forceinline__ void async_copy_b128(const float* gsrc, float* ldst) {
    unsigned int lds_addr = (unsigned int)(unsigned long long)(uintptr_t)ldst;
    unsigned long long gaddr = (unsigned long long)(uintptr_t)gsrc;
    asm volatile("global_load_async_to_lds_b128 %0, %1, off"
                 :: "v"(lds_addr), "v"(gaddr) : "memory");
}
__device__ __forceinline__ void wait_async_lds() {
    asm volatile("s_wait_asynccnt 0" ::: "memory");
}

// ---------------- kernel 1: one workgroup per query ------------------------
// Produces scores[q][b] = -SCALE * (2*ed_xy[q,b] - ed_q[q])
__global__ void energy_kernel(const float* __restrict__ anchors,
                              const float* __restrict__ cands,
                              const int*   __restrict__ amask,
                              float*       __restrict__ scores) {
    extern __shared__ float sm[];
    float* Al = sm + OFF_A;
    float* Cl = sm + OFF_C;
    float* Dl = sm + OFF_D;
    float* Ml = sm + OFF_M;
    float* Rl = sm + OFF_R;

    const int q    = blockIdx.x;
    const int tid  = threadIdx.x;
    const int nt   = blockDim.x;      // 512 = 16 wave32s
    const int lane = tid & 31;
    const int wv   = tid >> 5;

    // ---- Stage A-slice (128 KB) and candidates (64 KB) via async DMA ----
    const float* Ag = anchors + (size_t)q * Sn * En;
    for (int c = tid; c < Sn * (En / 4); c += nt) {     // 8192 b128 chunks
        int r = c >> 6, k = (c & 63) << 2;
        async_copy_b128(Ag + r * En + k, Al + r * SA + k);
    }
    for (int c = tid; c < Bn * (En / 4); c += nt) {     // 4096 b128 chunks
        int r = c >> 6, k = (c & 63) << 2;
        async_copy_b128(cands + r * En + k, Cl + r * SC + k);
    }
    if (tid < Sn) Ml[tid] = (float)amask[q * Sn + tid];
    wait_async_lds();
    __syncthreads();

    if (tid == 0) {                       // token count Σm (runs alongside)
        float s = 0.f;
        for (int i = 0; i < Sn; ++i) s += Ml[i];
        Rl[17] = s;
    }

    // ---- Phase 1: masked intra-query pairwise L1 (VALU; not bilinear) ----
    float acc = 0.f;
    for (int p = tid; p < Sn * Sn; p += nt) {
        int s1 = p >> 7, s2 = p & (Sn - 1);
        float mm = Ml[s1] * Ml[s2];
        if (mm != 0.f && s1 != s2) {      // diagonal contributes 0
            const float* r1 = Al + s1 * SA;
            const float* r2 = Al + s2 * SA;
            float d = 0.f;
#pragma unroll 8
            for (int e = 0; e < En; e += 4) {
                float4 x = *(const float4*)(r1 + e);
                float4 y = *(const float4*)(r2 + e);
                d += __builtin_fabsf(x.x - y.x) + __builtin_fabsf(x.y - y.y)
                   + __builtin_fabsf(x.z - y.z) + __builtin_fabsf(x.w - y.w);
            }
            acc += d;                      // mm == 1 for 0/1 mask
        }
    }
    for (int o = 16; o; o >>= 1) acc += __shfl_down(acc, o, 32);
    if (lane == 0) Rl[wv] = acc;
    __syncthreads();
    if (tid == 0) {
        float num = 0.f;
        for (int w = 0; w < nt / 32; ++w) num += Rl[w];
        float smv = Rl[17];
        Rl[16] = num / fmaxf(smv * smv, 1.0f);   // ed_q
    }

    // ---- Phase 2: anchor<->candidate L1 distances into LDS (raw) ----
    for (int p = tid; p < Sn * Bn; p += nt) {
        int s = p >> 6, b = p & (Bn - 1);
        float d = 0.f;
        if (Ml[s] != 0.f) {
            const float* r1 = Al + s * SA;
            const float* r2 = Cl + b * SC;
#pragma unroll 8
            for (int e = 0; e < En; e += 4) {
                float4 x = *(const float4*)(r1 + e);
                float4 y = *(const float4*)(r2 + e);
                d += __builtin_fabsf(x.x - y.x) + __builtin_fabsf(x.y - y.y)
                   + __builtin_fabsf(x.z - y.z) + __builtin_fabsf(x.w - y.w);
            }
        }
        Dl[s * SD + b] = d;
    }
    __syncthreads();

    // ---- Phase 3: ed_xy[b] = sum_s m[s]*dist[s,b] as f32 WMMA matvec ----
    // A(16x4) = mask broadcast down rows (VGPR0={K0|K2}, VGPR1={K1|K3}),
    // B(4x16) = dist rows (VGPR0={K0|K2} over N), C/D in f32 fragment layout.
    if (tid < 4 * 32) {                    // 4 waves, one 16-col b-tile each
        const int b0 = wv * 16;
        const int hi = lane >> 4, n = lane & 15;
        v8f accv = {0.f, 0.f, 0.f, 0.f, 0.f, 0.f, 0.f, 0.f};
        for (int k0 = 0; k0 < Sn; k0 += 4) {
            const int kA = k0 + (hi << 1);
            v2f av, bv;
            av.x = Ml[kA];
            av.y = Ml[kA + 1];
            bv.x = Dl[kA * SD + b0 + n];
            bv.y = Dl[(kA + 1) * SD + b0 + n];
            accv = __builtin_amdgcn_wmma_f32_16x16x4_f32(
                false, av, false, bv, (short)0, accv, false, false);
        }
        if (lane < 16) {                   // D row M=0 lives in VGPR0, lanes 0-15
            float vs     = fmaxf(Rl[17], 1.0f);
            float ed_xy  = accv[0] / vs;
            float energy = 2.0f * ed_xy - Rl[16];
            scores[q * Bn + b0 + lane] = -SCALE_F * energy;
        }
    }
}

// ---------------- kernel 2: diagonal-label softmax CE (1 wave) -------------
__global__ void ce_loss_kernel(const float* __restrict__ scores,
                               float* __restrict__ out) {
    const int q = threadIdx.x;             // 32 threads, one row each
    const float* row = scores + q * Bn;
    float mx = row[0];
    for (int b = 1; b < Bn; ++b) mx = fmaxf(mx, row[b]);
    float se = 0.f;
    for (int b = 0; b < Bn; ++b) se += expf(row[b] - mx);
    float lq = logf(se) + mx - row[q];     // logsumexp - score[q,q]
    for (int o = 16; o; o >>= 1) lq += __shfl_down(lq, o, 32);
    if (q == 0) out[0] = lq * (1.0f / Qn);
}

extern "C" void kernel_launch(void* const* d_in, const int* in_sizes, int n_in,
                              void* d_out, int out_size, void* d_ws, size_t ws_size,
                              hipStream_t stream) {
    const float* anchors = (const float*)d_in[0];   // [32,128,256] f32
    const float* cands   = (const float*)d_in[1];   // [64,256]     f32
    const int*   amask   = (const int*)d_in[2];     // [32,128]     i32
    float* scores = (float*)d_ws;                   // 32*64 f32 scratch
    float* out    = (float*)d_out;                  // scalar loss

    // >64KB dynamic LDS opt-in (idempotent; no stream interaction)
    (void)hipFuncSetAttribute((const void*)energy_kernel,
                              hipFuncAttributeMaxDynamicSharedMemorySize,
                              SMEM_BYTES);

    energy_kernel<<<Qn, 512, SMEM_BYTES, stream>>>(anchors, cands, amask, scores);
    ce_loss_kernel<<<1, 32, 0, stream>>>(scores, out);
}